// DimXsimCL_Encoder_27676769255725
// MI455X (gfx1250) — compile-verified
//
#include <hip/hip_runtime.h>
#include <hip/hip_bf16.h>

// LightGCN-style propagation for MI455X (gfx1250, wave32).
//  - SpMM is memory/atomic bound; node matrix (51.2MB) is L2-resident (192MB L2),
//    so we use per-edge fp32 atomics (global_atomic_add_f32) + b128 gathers.
//  - The per-layer "acc += ego * (1/3)" accumulate is done on the matrix pipe as
//    D = A x (I/3) + C with V_WMMA_F32_16X16X4_F32 (exact fp32 arithmetic).

typedef __attribute__((ext_vector_type(2))) float v2f;
typedef __attribute__((ext_vector_type(8))) float v8f;

#define EMB 64

// ---------------------------------------------------------------- zero buffer
__global__ void zero_f4(float4* __restrict__ p, int n4) {
    int i = blockIdx.x * blockDim.x + threadIdx.x;
    if (i < n4) p[i] = make_float4(0.f, 0.f, 0.f, 0.f);
}

// ------------------------------------------------- concat(user, item) -> dst
__global__ void concat_f4(const float4* __restrict__ user,
                          const float4* __restrict__ item,
                          float4* __restrict__ dst,
                          int user_n4, int total_n4) {
    int i = blockIdx.x * blockDim.x + threadIdx.x;
    if (i >= total_n4) return;
    dst[i] = (i < user_n4) ? user[i] : item[i - user_n4];
}

// ---------------------------------------------------------------- COO SpMM
// 16 lanes per edge, each lane owns a float4 (16B) slice of the 256B row.
// y[row[e]] += val[e] * x[col[e]]  via hardware fp32 atomics (L2-resident).
__global__ void spmm_coo(const float* __restrict__ x,
                         float* __restrict__ y,
                         const int* __restrict__ row,
                         const int* __restrict__ col,
                         const float* __restrict__ val,
                         int nnz) {
    int t = blockIdx.x * blockDim.x + threadIdx.x;
    int e = t >> 4;          // edge index
    int q = t & 15;          // float4 slice within the 64-float row
    if (e >= nnz) return;
    int r = row[e];
    int c = col[e];
    float v = val[e];
    const float4 xv = *(const float4*)(x + (size_t)c * EMB + q * 4);
    float* yp = y + (size_t)r * EMB + q * 4;
    __hip_atomic_fetch_add(yp + 0, v * xv.x, __ATOMIC_RELAXED, __HIP_MEMORY_SCOPE_AGENT);
    __hip_atomic_fetch_add(yp + 1, v * xv.y, __ATOMIC_RELAXED, __HIP_MEMORY_SCOPE_AGENT);
    __hip_atomic_fetch_add(yp + 2, v * xv.z, __ATOMIC_RELAXED, __HIP_MEMORY_SCOPE_AGENT);
    __hip_atomic_fetch_add(yp + 3, v * xv.w, __ATOMIC_RELAXED, __HIP_MEMORY_SCOPE_AGENT);
}

// --------------------------------------------- acc += ego * scale, via WMMA
// One wave per 16x16 tile of the [N, 64] matrix. The 16x16 scaled identity is
// split into four K=4 slabs: acc_tile = sum_j A_j x B_j + acc_tile, where
// A_j = ego tile columns [4j..4j+3] (16x4) and B_j[k,n] = scale * (n == 4j+k).
//
// f32 WMMA operand layouts (CDNA5 ISA 7.12.2):
//   A 16x4 : lanes 0-15 -> M=lane, VGPR{0,1} = K{0,1}; lanes 16-31 -> K{2,3}
//   B 4x16 : lanes 0-15 -> N=lane, VGPR{0,1} = K{0,1}; lanes 16-31 -> K{2,3}
//   C/D    : VGPR r, lanes 0-15 -> (M=r, N=lane); lanes 16-31 -> (M=r+8, N=lane-16)
__global__ void acc_scale_wmma(const float* __restrict__ ego,
                               float* __restrict__ acc,
                               int n_tiles, float scale) {
    int wave = blockIdx.x * (blockDim.x >> 5) + (threadIdx.x >> 5);
    if (wave >= n_tiles) return;              // wave-uniform: EXEC stays all-1s
    int lane = threadIdx.x & 31;
    int half = lane >> 4;                     // 0: lanes 0-15, 1: lanes 16-31
    int lm   = lane & 15;

    int m0 = (wave >> 2) << 4;                // row-tile (node dim)
    int n0 = (wave & 3) << 4;                 // col-tile (4 tiles across EMB=64)

    // Load C (accumulator tile) in WMMA C/D layout.
    float* cbase = acc + (size_t)(m0 + 8 * half) * EMB + n0 + lm;
    v8f c;
#pragma unroll
    for (int r = 0; r < 8; ++r) c[r] = cbase[r * EMB];

#pragma unroll
    for (int j = 0; j < 4; ++j) {
        // A_j: this lane holds ego[m0+lm, n0+4j+2*half .. +1]
        const float* ap = ego + (size_t)(m0 + lm) * EMB + n0 + 4 * j + 2 * half;
        v2f a;
        a.x = ap[0];
        a.y = ap[1];
        // B_j: scaled identity slice; this lane holds column n=lm, K=2*half..+1
        int kb = 4 * j + 2 * half;
        v2f b;
        b.x = (kb     == lm) ? scale : 0.0f;
        b.y = (kb + 1 == lm) ? scale : 0.0f;
        c = __builtin_amdgcn_wmma_f32_16x16x4_f32(
                /*neg_a=*/false, a, /*neg_b=*/false, b,
                /*c_mod=*/(short)0, c, /*reuse_a=*/false, /*reuse_b=*/false);
    }

#pragma unroll
    for (int r = 0; r < 8; ++r) cbase[r * EMB] = c[r];
}

extern "C" void kernel_launch(void* const* d_in, const int* in_sizes, int n_in,
                              void* d_out, int out_size, void* d_ws, size_t ws_size,
                              hipStream_t stream) {
    const float* user = (const float*)d_in[0];
    const float* item = (const float*)d_in[1];
    const int*   row  = (const int*)d_in[2];
    const int*   col  = (const int*)d_in[3];
    const float* val  = (const float*)d_in[4];

    const int user_elems = in_sizes[0];
    const int item_elems = in_sizes[1];
    const int nnz        = in_sizes[2];
    const int N          = (user_elems + item_elems) / EMB;

    float* acc = (float*)d_out;
    float* W0  = (float*)d_ws;                       // ego (layer input)
    float* W1  = W0 + (size_t)N * EMB;               // spmm output

    const size_t elems = (size_t)N * EMB;
    const int n4 = (int)(elems / 4);
    const int threads = 256;

    // ego0 = concat(user_emb, item_emb)
    concat_f4<<<(n4 + threads - 1) / threads, threads, 0, stream>>>(
        (const float4*)user, (const float4*)item, (float4*)W0,
        user_elems / 4, n4);

    // acc = 0
    zero_f4<<<(n4 + threads - 1) / threads, threads, 0, stream>>>((float4*)acc, n4);

    const int n_tiles    = (N / 16) * (EMB / 16);            // 16x16 tiles
    const int wmma_blks  = (n_tiles + (threads / 32) - 1) / (threads / 32);
    const long long spmm_threads = (long long)nnz * 16;
    const int spmm_blks  = (int)((spmm_threads + threads - 1) / threads);

    for (int layer = 0; layer < 3; ++layer) {
        // W1 = 0
        zero_f4<<<(n4 + threads - 1) / threads, threads, 0, stream>>>((float4*)W1, n4);
        // W1 += A * W0   (COO scatter-add SpMM)
        spmm_coo<<<spmm_blks, threads, 0, stream>>>(W0, W1, row, col, val, nnz);
        // acc += W1 * (1/3)   via v_wmma_f32_16x16x4_f32
        acc_scale_wmma<<<wmma_blks, threads, 0, stream>>>(W1, acc, n_tiles, 1.0f / 3.0f);
        // swap ping-pong buffers
        float* t = W0; W0 = W1; W1 = t;
    }
}